// SimpleModel_75634374082958
// MI455X (gfx1250) — compile-verified
//
#include <hip/hip_runtime.h>
#include <hip/hip_bf16.h>
#include <math.h>

#define V_ 32000
#define D_ 256
#define H_ 8
#define S_ 1024
#define B_ 2

typedef __attribute__((ext_vector_type(2))) float v2f;
typedef __attribute__((ext_vector_type(8))) float v8f;
typedef unsigned int u32;
typedef u32 u32x4 __attribute__((ext_vector_type(4)));
typedef u32 u32x8 __attribute__((ext_vector_type(8)));

__device__ __forceinline__ v8f wmma4(v2f a, v2f b, v8f c) {
  // D = A(16x4) * B(4x16) + C, f32 in / f32 accumulate
  return __builtin_amdgcn_wmma_f32_16x16x4_f32(
      /*neg_a=*/false, a, /*neg_b=*/false, b,
      /*c_mod=*/(short)0, c, /*reuse_a=*/false, /*reuse_b=*/false);
}

// ---------------------------------------------------------------------------
// 1) x = embed[tokens]; xn = rmsnorm(x, rms1_scale)
// ---------------------------------------------------------------------------
__global__ void embed_rms_kernel(const int* __restrict__ tokens,
                                 const float* __restrict__ embed,
                                 const float* __restrict__ scale,
                                 float* __restrict__ xn) {
  int row = blockIdx.x;          // 0 .. B*S-1
  int tid = threadIdx.x;         // 0 .. 255
  __shared__ float red[256];
  int tok = tokens[row];
  float val = embed[(size_t)tok * D_ + tid];
  red[tid] = val * val;
  __syncthreads();
  for (int s = 128; s > 0; s >>= 1) {
    if (tid < s) red[tid] += red[tid + s];
    __syncthreads();
  }
  float inv = rsqrtf(red[0] * (1.0f / (float)D_) + 1e-6f);
  xn[(size_t)row * D_ + tid] = val * inv * scale[tid];
}

// ---------------------------------------------------------------------------
// 2) q/k/v = xn @ W{q,k,v}[h] + b{q,k,v}[h]   (WMMA, A-fragment shared 3x)
// ---------------------------------------------------------------------------
__global__ void qkv_kernel(const float* __restrict__ xn,
                           const float* __restrict__ Wq, const float* __restrict__ bq,
                           const float* __restrict__ Wk, const float* __restrict__ bk,
                           const float* __restrict__ Wv, const float* __restrict__ bv,
                           float* __restrict__ q, float* __restrict__ k,
                           float* __restrict__ v) {
  int wid = blockIdx.x * (blockDim.x >> 5) + (threadIdx.x >> 5);
  int et = wid & 15;             // D/16 tiles
  int st = (wid >> 4) & 63;      // S/16 tiles
  int h  = (wid >> 10) & 7;
  int b  = wid >> 13;
  int lane = threadIdx.x & 31;
  int hf = lane >> 4, l = lane & 15;
  int e0 = et * 16, s0 = st * 16;

  const float* A  = xn + (size_t)b * S_ * D_;
  const float* WQ = Wq + (size_t)h * D_ * D_;
  const float* WK = Wk + (size_t)h * D_ * D_;
  const float* WV = Wv + (size_t)h * D_ * D_;

  float biq = bq[h * D_ + e0 + l];
  float bik = bk[h * D_ + e0 + l];
  float biv = bv[h * D_ + e0 + l];
  v8f aq, ak, av;
  for (int r = 0; r < 8; ++r) { aq[r] = biq; ak[r] = bik; av[r] = biv; }

  for (int k0 = 0; k0 < D_; k0 += 4) {
    int kk = k0 + 2 * hf;
    v2f afr = *(const v2f*)&A[(size_t)(s0 + l) * D_ + kk];
    v2f bfq; bfq.x = WQ[(size_t)kk * D_ + e0 + l]; bfq.y = WQ[(size_t)(kk + 1) * D_ + e0 + l];
    v2f bfk; bfk.x = WK[(size_t)kk * D_ + e0 + l]; bfk.y = WK[(size_t)(kk + 1) * D_ + e0 + l];
    v2f bfv; bfv.x = WV[(size_t)kk * D_ + e0 + l]; bfv.y = WV[(size_t)(kk + 1) * D_ + e0 + l];
    aq = wmma4(afr, bfq, aq);
    ak = wmma4(afr, bfk, ak);
    av = wmma4(afr, bfv, av);
  }

  size_t base = (size_t)(b * H_ + h) * S_ * D_;
  for (int r = 0; r < 8; ++r) {
    int s = s0 + r + 8 * hf;
    size_t o = base + (size_t)s * D_ + e0 + l;
    q[o] = aq[r]; k[o] = ak[r]; v[o] = av[r];
  }
}

// ---------------------------------------------------------------------------
// 3) RoPE on q, k, v (reference applies rope to v too; (i-1) theta quirk kept)
// ---------------------------------------------------------------------------
__global__ void rope_kernel(float* __restrict__ q, float* __restrict__ k,
                            float* __restrict__ v) {
  const long long PP = (long long)B_ * H_ * S_ * (D_ / 2);
  long long idx = (long long)blockIdx.x * blockDim.x + threadIdx.x;
  if (idx >= 3 * PP) return;
  float* ptr = (idx < PP) ? q : ((idx < 2 * PP) ? k : v);
  long long rem = idx % PP;
  int p = (int)(rem & 127);                 // pair index 0..127
  long long row = rem >> 7;                 // (b*H+h)*S + s
  int s = (int)(row & (S_ - 1));
  float theta = powf(10000.0f, -2.0f * ((float)p - 1.0f) / (float)D_);
  float ang = (float)s * theta;
  float sn, cs;
  sincosf(ang, &sn, &cs);
  float* base = ptr + row * D_ + 2 * p;
  float xe = base[0], xo = base[1];
  base[0] =  cs * xe + sn * xo;
  base[1] = -sn * xe + cs * xo;
}

// ---------------------------------------------------------------------------
// 4) Causal attention, one wave per 16-row query tile per (b,h).
//    Scores staged in 64KB LDS, softmax in-wave, WMMA for Q*K^T and P*V.
// ---------------------------------------------------------------------------
__global__ void attn_kernel(const float* __restrict__ q, const float* __restrict__ k,
                            const float* __restrict__ v, float* __restrict__ heads) {
  __shared__ float P[16 * S_];   // 65536 bytes (<= 320KB/WGP)
  int blk = blockIdx.x;
  int qt = blk % (S_ / 16);
  int bh = blk / (S_ / 16);
  const float* Q = q + (size_t)bh * S_ * D_;
  const float* K = k + (size_t)bh * S_ * D_;
  const float* Vv = v + (size_t)bh * S_ * D_;
  int lane = threadIdx.x & 31;
  int hf = lane >> 4, l = lane & 15;
  int q0 = qt * 16;

  for (int j = 0; j <= qt; ++j) {
    int t0 = j * 16;
    v8f acc = {};
    for (int k0 = 0; k0 < D_; k0 += 4) {
      int kk = k0 + 2 * hf;
      v2f a = *(const v2f*)&Q[(size_t)(q0 + l) * D_ + kk];
      v2f b = *(const v2f*)&K[(size_t)(t0 + l) * D_ + kk];   // B = K^T fragment
      acc = wmma4(a, b, acc);
    }
    for (int r = 0; r < 8; ++r) {
      int row = r + 8 * hf;           // 0..15 within tile
      int sq = q0 + row;
      int t = t0 + l;
      float sc = acc[r] * 0.0625f;    // 1/sqrt(256)
      if (t > sq) sc = -1e9f;
      P[row * S_ + t] = sc;
    }
  }
  __syncthreads();

  int nk = (qt + 1) * 16;
  if (lane < 16) {
    int row = lane;
    float mx = -1e30f;
    for (int t = 0; t < nk; ++t) mx = fmaxf(mx, P[row * S_ + t]);
    float sum = 0.0f;
    for (int t = 0; t < nk; ++t) {
      float e = __expf(P[row * S_ + t] - mx);
      P[row * S_ + t] = e;
      sum += e;
    }
    float inv = 1.0f / sum;
    for (int t = 0; t < nk; ++t) P[row * S_ + t] *= inv;
  }
  __syncthreads();

  int b_ = bh / H_, h_ = bh % H_;
  for (int n0 = 0; n0 < D_; n0 += 16) {
    v8f acc = {};
    for (int k0 = 0; k0 < nk; k0 += 4) {
      int kk = k0 + 2 * hf;
      v2f a = *(const v2f*)&P[l * S_ + kk];
      v2f b;
      b.x = Vv[(size_t)kk * D_ + n0 + l];
      b.y = Vv[(size_t)(kk + 1) * D_ + n0 + l];
      acc = wmma4(a, b, acc);
    }
    for (int r = 0; r < 8; ++r) {
      int s = q0 + r + 8 * hf;
      heads[((size_t)(b_ * S_ + s)) * (H_ * D_) + h_ * D_ + n0 + l] = acc[r];
    }
  }
}

// ---------------------------------------------------------------------------
// 5) y = xn + heads @ Wo + bo     (M=2048, K=2048, N=256)
// ---------------------------------------------------------------------------
__global__ void oproj_kernel(const float* __restrict__ heads,
                             const float* __restrict__ Wo,
                             const float* __restrict__ bo,
                             const float* __restrict__ xn,
                             float* __restrict__ y) {
  int wid = blockIdx.x * (blockDim.x >> 5) + (threadIdx.x >> 5);
  int nt = wid & 15;          // N/16 = 16
  int mt = wid >> 4;          // M/16 = 128
  int lane = threadIdx.x & 31;
  int hf = lane >> 4, l = lane & 15;
  int m0 = mt * 16, n0 = nt * 16;
  const int Kd = H_ * D_;

  float bias = bo[n0 + l];
  v8f acc;
  for (int r = 0; r < 8; ++r) acc[r] = bias;
  for (int k0 = 0; k0 < Kd; k0 += 4) {
    int kk = k0 + 2 * hf;
    v2f a = *(const v2f*)&heads[(size_t)(m0 + l) * Kd + kk];
    v2f b; b.x = Wo[(size_t)kk * D_ + n0 + l]; b.y = Wo[(size_t)(kk + 1) * D_ + n0 + l];
    acc = wmma4(a, b, acc);
  }
  for (int r = 0; r < 8; ++r) {
    int row = m0 + r + 8 * hf;
    y[(size_t)row * D_ + n0 + l] = acc[r] + xn[(size_t)row * D_ + n0 + l];
  }
}

// ---------------------------------------------------------------------------
// 6) t = rmsnorm(y, rms2_scale); x2 = t + relu(t)
// ---------------------------------------------------------------------------
__global__ void rms2_kernel(const float* __restrict__ y,
                            const float* __restrict__ scale,
                            float* __restrict__ x2) {
  int row = blockIdx.x;
  int tid = threadIdx.x;
  __shared__ float red[256];
  float val = y[(size_t)row * D_ + tid];
  red[tid] = val * val;
  __syncthreads();
  for (int s = 128; s > 0; s >>= 1) {
    if (tid < s) red[tid] += red[tid + s];
    __syncthreads();
  }
  float inv = rsqrtf(red[0] * (1.0f / (float)D_) + 1e-6f);
  float t = val * inv * scale[tid];
  x2[(size_t)row * D_ + tid] = t + fmaxf(t, 0.0f);
}

// ---------------------------------------------------------------------------
// 7) logits = x2 @ Wout + bout    (M=2048, K=256, N=32000) -> d_out
//    Block = 8 waves = 8 n-tiles sharing one 16x256 A tile staged in LDS by
//    the Tensor Data Mover (wave 0 issues tensor_load_to_lds, waits on
//    TENSORcnt, then the workgroup barrier publishes the tile).
// ---------------------------------------------------------------------------
__global__ void logits_kernel(const float* __restrict__ x2,
                              const float* __restrict__ Wout,
                              const float* __restrict__ bout,
                              float* __restrict__ out) {
  __shared__ float As[16 * D_];            // 16 KB staged A tile
  const int NG = (V_ / 16) / 8;            // 250 n-tile groups per m-tile
  int mt = blockIdx.x / NG;                // 0..127
  int ng = blockIdx.x % NG;                // 0..249
  int wave = threadIdx.x >> 5;
  int nt = ng * 8 + wave;
  int lane = threadIdx.x & 31;
  int hf = lane >> 4, l = lane & 15;
  int m0 = mt * 16, n0 = nt * 16;

  if (threadIdx.x < 32) {
    // ---- TDM: 2D tile, 16 rows x 256 f32, row stride 256 elements ----
    unsigned long long ga =
        (unsigned long long)(const void*)(x2 + (size_t)m0 * D_);
    u32 lds_base = (u32)(unsigned long long)(void*)(&As[0]);  // LDS byte offset
    u32x4 g0;
    g0[0] = 1u;                                    // count=1 (user descriptor)
    g0[1] = lds_base;                              // lds_addr
    g0[2] = (u32)(ga & 0xffffffffu);               // global_addr[31:0]
    g0[3] = (u32)((ga >> 32) & 0x01ffffffu)        // global_addr[56:32]
            | (2u << 30);                          // type = 2 ("image")
    u32x8 g1;
    g1[0] = 2u << 16;                              // data_size = 4 bytes
    g1[1] = (u32)D_ << 16;                         // tensor_dim0 = 256
    g1[2] = 16u << 16;                             // tensor_dim1 = 16
    g1[3] = (u32)D_ << 16;                         // tile_dim0   = 256
    g1[4] = 16u;                                   // tile_dim1   = 16, tile_dim2 = 0
    g1[5] = (u32)D_;                               // tensor_dim0_stride = 256
    g1[6] = 0u;                                    // stride hi / dim1_stride lo
    g1[7] = 0u;                                    // dim1_stride hi
    asm volatile("tensor_load_to_lds %0, %1" :: "s"(g0), "s"(g1) : "memory");
    __builtin_amdgcn_s_wait_tensorcnt(0);
  }
  __syncthreads();

  const float* Bp = Wout + n0 + l;
  __builtin_prefetch(Bp, 0, 1);                    // global_prefetch_b8

  float bias = bout[n0 + l];
  v8f acc;
  for (int r = 0; r < 8; ++r) acc[r] = bias;
  for (int k0 = 0; k0 < D_; k0 += 4) {
    int kk = k0 + 2 * hf;
    v2f a = *(const v2f*)&As[(size_t)l * D_ + kk];  // A-fragment from LDS
    v2f b; b.x = Bp[(size_t)kk * V_]; b.y = Bp[(size_t)(kk + 1) * V_];
    acc = wmma4(a, b, acc);
  }
  for (int r = 0; r < 8; ++r) {
    int row = m0 + r + 8 * hf;
    out[(size_t)row * V_ + n0 + l] = acc[r];
  }
}

// ---------------------------------------------------------------------------
extern "C" void kernel_launch(void* const* d_in, const int* in_sizes, int n_in,
                              void* d_out, int out_size, void* d_ws, size_t ws_size,
                              hipStream_t stream) {
  (void)in_sizes; (void)n_in; (void)out_size; (void)ws_size;
  const int*   tokens = (const int*)d_in[0];
  const float* embed  = (const float*)d_in[1];
  const float* rms1   = (const float*)d_in[2];
  const float* Wq     = (const float*)d_in[3];
  const float* bq     = (const float*)d_in[4];
  const float* Wk     = (const float*)d_in[5];
  const float* bk     = (const float*)d_in[6];
  const float* Wv     = (const float*)d_in[7];
  const float* bv     = (const float*)d_in[8];
  const float* Wo     = (const float*)d_in[9];
  const float* bo     = (const float*)d_in[10];
  const float* rms2   = (const float*)d_in[11];
  const float* Wout   = (const float*)d_in[12];
  const float* bout   = (const float*)d_in[13];
  float* out = (float*)d_out;

  float* ws = (float*)d_ws;
  float* xn    = ws;                                  // B*S*D
  float* q     = xn + (size_t)B_ * S_ * D_;           // B*H*S*D
  float* k     = q  + (size_t)B_ * H_ * S_ * D_;
  float* v     = k  + (size_t)B_ * H_ * S_ * D_;
  float* heads = v  + (size_t)B_ * H_ * S_ * D_;      // B*S*H*D
  float* y     = heads + (size_t)B_ * S_ * H_ * D_;   // B*S*D
  float* x2    = y  + (size_t)B_ * S_ * D_;           // B*S*D

  embed_rms_kernel<<<B_ * S_, 256, 0, stream>>>(tokens, embed, rms1, xn);

  qkv_kernel<<<2048, 256, 0, stream>>>(xn, Wq, bq, Wk, bk, Wv, bv, q, k, v);

  long long npairs = 3LL * B_ * H_ * S_ * (D_ / 2);
  rope_kernel<<<(int)((npairs + 255) / 256), 256, 0, stream>>>(q, k, v);

  attn_kernel<<<B_ * H_ * (S_ / 16), 32, 0, stream>>>(q, k, v, heads);

  oproj_kernel<<<256, 256, 0, stream>>>(heads, Wo, bo, xn, y);

  rms2_kernel<<<B_ * S_, 256, 0, stream>>>(y, rms2, x2);

  logits_kernel<<<32000, 256, 0, stream>>>(x2, Wout, bout, out);
}